// PolymerGCN_69286412419642
// MI455X (gfx1250) — compile-verified
//
#include <hip/hip_runtime.h>
#include <hip/hip_bf16.h>

// ---------------------------------------------------------------------------
// PolymerGCN forward: 8x (GCNConv -> BN(eval) -> ReLU) -> mean-pool -> linear
// N=50000 nodes, E=640000 edges, D=128, L=8, G=2048 graphs, OUT=5.
// GEMMs: exact-fp32 V_WMMA_F32_16X16X4_F32, 80-row blocks, 5x register
// blocking per wave; tile staging via GLOBAL_LOAD_ASYNC_TO_LDS_B128.
// ---------------------------------------------------------------------------

#define GCN_N   50000
#define GCN_E   640000
#define GCN_D   128
#define GCN_L   8
#define GCN_G   2048
#define GCN_OUT 5
#define BN_EPS  1e-5f

#define M_TILE  80              // rows per block (50000 = 625 * 80)
#define M_SUB   5               // 16-row sub-tiles per wave
#define LDS_STRIDE 132          // padded row stride (floats)

typedef __attribute__((ext_vector_type(2))) float v2f;
typedef __attribute__((ext_vector_type(8))) float v8f;
typedef __attribute__((address_space(3))) float* lds_fp;

// ------------------------------- utility -----------------------------------

__global__ __launch_bounds__(256) void zero_kernel(float* __restrict__ p, int n) {
    int i = blockIdx.x * 256 + threadIdx.x;
    if (i < n) p[i] = 0.0f;
}

// deg[dst] += 1 for every edge (float atomics; exact for counts < 2^24)
__global__ __launch_bounds__(256) void count_deg_kernel(const int* __restrict__ ei,
                                                        float* __restrict__ deg, int nE) {
    int e = blockIdx.x * 256 + threadIdx.x;
    if (e < nE) atomicAdd(&deg[ei[nE + e]], 1.0f);
}

// in-place: dis[n] = rsqrt(deg[n] + 1)
__global__ __launch_bounds__(256) void rsqrt_deg_kernel(float* __restrict__ d, int nN) {
    int n = blockIdx.x * 256 + threadIdx.x;
    if (n < nN) d[n] = rsqrtf(d[n] + 1.0f);
}

// ------------------------- WMMA fp32 GEMM: t = h @ W ------------------------
// One block = 80 rows of h staged in LDS (async). 8 waves; wave w owns output
// columns [16w,16w+16) and keeps 5 accumulators (one per 16-row sub-tile), so
// each B fragment of W is reused 5x. K-loop: 32 steps of
// V_WMMA_F32_16X16X4_F32 per accumulator (exact fp32 MACs).
//
// Fragment layouts (ISA 7.12.2, wave32):
//   A 16x4 : lane L (L<16): M=L, {VGPR0,VGPR1} = K{0,1}; lanes 16-31: K{2,3}
//   B 4x16 : lane L (L<16): N=L, {VGPR0,VGPR1} = rows K{0,1}; lanes 16-31: K{2,3}
//   C 16x16: VGPR r: lanes 0-15 -> (M=r, N=lane); lanes 16-31 -> (M=r+8, N=lane-16)
__global__ __launch_bounds__(256) void gemm_wmma_kernel(const float* __restrict__ hin,
                                                        const float* __restrict__ Wl,
                                                        float* __restrict__ tout) {
    __shared__ float lds[M_TILE * LDS_STRIDE];    // 80 x 128 tile, padded
    const int rb  = blockIdx.x * M_TILE;
    const int tid = threadIdx.x;

    // async-stage 80x128 fp32 tile: 2560 b128 transfers, 10 per thread,
    // no VGPR round-trip (ASYNCcnt-tracked direct global->LDS DMA).
    const unsigned ldsBase = (unsigned)(uintptr_t)(lds_fp)(&lds[0]);
    #pragma unroll
    for (int i = tid; i < M_TILE * 32; i += 256) {
        const int r  = i >> 5;                    // 0..79
        const int c4 = i & 31;                    // float4 column 0..31
        const unsigned lof = ldsBase + (unsigned)(r * LDS_STRIDE + c4 * 4) * 4u;
        const unsigned long long gaddr =
            (unsigned long long)(uintptr_t)(hin + (size_t)(rb + r) * GCN_D + c4 * 4);
        asm volatile("global_load_async_to_lds_b128 %0, %1, off"
                     :: "v"(lof), "v"(gaddr) : "memory");
    }
    asm volatile("s_wait_asynccnt 0x0" ::: "memory");
    __syncthreads();

    const int wave  = tid >> 5;                   // 0..7 -> column tile
    const int lane  = tid & 31;
    const int cb    = wave * 16;
    const int m     = lane & 15;                  // A row / B,C column index
    const int khalf = (lane < 16) ? 0 : 2;        // K sub-group per lane half

    v8f acc[M_SUB];
    #pragma unroll
    for (int s = 0; s < M_SUB; ++s) acc[s] = (v8f){};

    for (int k0 = 0; k0 < GCN_D; k0 += 4) {
        v2f b;                                    // shared across the 5 sub-tiles
        b.x = Wl[(k0 + khalf)     * GCN_D + cb + m];
        b.y = Wl[(k0 + khalf + 1) * GCN_D + cb + m];
        #pragma unroll
        for (int s = 0; s < M_SUB; ++s) {
            v2f a;
            a.x = lds[(s * 16 + m) * LDS_STRIDE + k0 + khalf];
            a.y = lds[(s * 16 + m) * LDS_STRIDE + k0 + khalf + 1];
            acc[s] = __builtin_amdgcn_wmma_f32_16x16x4_f32(
                /*neg_a=*/false, a, /*neg_b=*/false, b,
                /*c_mod=*/(short)0, acc[s], /*reuse_a=*/false, /*reuse_b=*/false);
        }
    }

    const int mbase = (lane < 16) ? 0 : 8;
    #pragma unroll
    for (int s = 0; s < M_SUB; ++s)
        #pragma unroll
        for (int r = 0; r < 8; ++r)
            tout[(size_t)(rb + s * 16 + mbase + r) * GCN_D + cb + m] = acc[s][r];
}

// ---------------- agg = t * self_norm + bias (fully overwrites agg) ---------
__global__ __launch_bounds__(256) void init_self_kernel(const float* __restrict__ t,
                                                        const float* __restrict__ dis,
                                                        const float* __restrict__ bl,
                                                        float* __restrict__ agg, int nN) {
    const int gid = blockIdx.x * 256 + threadIdx.x;     // nN*32 threads
    const int n = gid >> 5;
    if (n >= nN) return;
    const int f = (gid & 31) * 4;
    const float s = dis[n] * dis[n];                    // = 1/deg
    const float4 v = *(const float4*)(t + (size_t)n * GCN_D + f);
    float4 o;
    o.x = v.x * s + bl[f + 0];
    o.y = v.y * s + bl[f + 1];
    o.z = v.z * s + bl[f + 2];
    o.w = v.w * s + bl[f + 3];
    *(float4*)(agg + (size_t)n * GCN_D + f) = o;
}

// ---------------- agg[dst] += t[src] * dis[src]*dis[dst] --------------------
__global__ __launch_bounds__(256) void edge_scatter_kernel(const int* __restrict__ ei,
                                                           const float* __restrict__ t,
                                                           const float* __restrict__ dis,
                                                           float* __restrict__ agg, int nE) {
    const int gid = blockIdx.x * 256 + threadIdx.x;     // nE*32 threads
    const int e = gid >> 5;
    if (e >= nE) return;
    const int f   = (gid & 31) * 4;
    const int src = ei[e];
    const int dst = ei[nE + e];
    const float w = dis[src] * dis[dst];
    const float4 v = *(const float4*)(t + (size_t)src * GCN_D + f);
    float* o = agg + (size_t)dst * GCN_D + f;
    atomicAdd(o + 0, v.x * w);
    atomicAdd(o + 1, v.y * w);
    atomicAdd(o + 2, v.z * w);
    atomicAdd(o + 3, v.w * w);
}

// ---------------- h = relu(bn_eval(agg)) ------------------------------------
__global__ __launch_bounds__(256) void bn_relu_kernel(const float* __restrict__ agg,
                                                      const float* __restrict__ gam,
                                                      const float* __restrict__ bet,
                                                      const float* __restrict__ rm,
                                                      const float* __restrict__ rv,
                                                      float* __restrict__ h, int nN) {
    const int gid = blockIdx.x * 256 + threadIdx.x;
    const int n = gid >> 5;
    if (n >= nN) return;
    const int f = (gid & 31) * 4;
    const float4 a = *(const float4*)(agg + (size_t)n * GCN_D + f);
    float4 o;
    o.x = fmaxf((a.x - rm[f + 0]) * rsqrtf(rv[f + 0] + BN_EPS) * gam[f + 0] + bet[f + 0], 0.0f);
    o.y = fmaxf((a.y - rm[f + 1]) * rsqrtf(rv[f + 1] + BN_EPS) * gam[f + 1] + bet[f + 1], 0.0f);
    o.z = fmaxf((a.z - rm[f + 2]) * rsqrtf(rv[f + 2] + BN_EPS) * gam[f + 2] + bet[f + 2], 0.0f);
    o.w = fmaxf((a.w - rm[f + 3]) * rsqrtf(rv[f + 3] + BN_EPS) * gam[f + 3] + bet[f + 3], 0.0f);
    *(float4*)(h + (size_t)n * GCN_D + f) = o;
}

// ---------------- pooled[batch[n]] += h[n]; cnt[batch[n]] += 1 --------------
__global__ __launch_bounds__(256) void pool_add_kernel(const float* __restrict__ h,
                                                       const int* __restrict__ batch,
                                                       float* __restrict__ pooled,
                                                       float* __restrict__ cnt, int nN) {
    const int gid = blockIdx.x * 256 + threadIdx.x;
    const int n = gid >> 5;
    if (n >= nN) return;
    const int f = (gid & 31) * 4;
    const int g = batch[n];
    const float4 v = *(const float4*)(h + (size_t)n * GCN_D + f);
    float* o = pooled + (size_t)g * GCN_D + f;
    atomicAdd(o + 0, v.x);
    atomicAdd(o + 1, v.y);
    atomicAdd(o + 2, v.z);
    atomicAdd(o + 3, v.w);
    if (f == 0) atomicAdd(&cnt[g], 1.0f);
}

// ---------------- out[g][o] = (pooled[g]/max(cnt,1)) @ W_out + b_out --------
__global__ __launch_bounds__(256) void out_gemm_kernel(const float* __restrict__ pooled,
                                                       const float* __restrict__ cnt,
                                                       const float* __restrict__ Wout,
                                                       const float* __restrict__ bout,
                                                       float* __restrict__ out, int total) {
    const int gid = blockIdx.x * 256 + threadIdx.x;
    if (gid >= total) return;
    const int g = gid / GCN_OUT;
    const int o = gid % GCN_OUT;
    const float inv = 1.0f / fmaxf(cnt[g], 1.0f);
    float s = 0.0f;
    #pragma unroll 8
    for (int d = 0; d < GCN_D; ++d)
        s += pooled[(size_t)g * GCN_D + d] * Wout[d * GCN_OUT + o];
    out[gid] = s * inv + bout[o];
}

// ---------------------------------------------------------------------------

extern "C" void kernel_launch(void* const* d_in, const int* in_sizes, int n_in,
                              void* d_out, int out_size, void* d_ws, size_t ws_size,
                              hipStream_t stream) {
    (void)in_sizes; (void)n_in; (void)out_size; (void)ws_size;

    const float* x      = (const float*)d_in[0];   // [N,D]
    const int*   ei     = (const int*)  d_in[1];   // [2,E]
    const int*   batch  = (const int*)  d_in[2];   // [N]
    const float* W      = (const float*)d_in[3];   // [L,D,D]
    const float* b      = (const float*)d_in[4];   // [L,D]
    const float* gamma  = (const float*)d_in[5];   // [L,D]
    const float* beta   = (const float*)d_in[6];   // [L,D]
    const float* rmean  = (const float*)d_in[7];   // [L,D]
    const float* rvar   = (const float*)d_in[8];   // [L,D]
    const float* W_out  = (const float*)d_in[9];   // [D,OUT]
    const float* b_out  = (const float*)d_in[10];  // [OUT]
    float*       out    = (float*)d_out;           // [G,OUT]

    // workspace layout (floats, all 16B aligned)
    float* ws = (float*)d_ws;
    const size_t ND = (size_t)GCN_N * GCN_D;
    float* t      = ws;                         // [N,D]
    float* agg    = t + ND;                     // [N,D]
    float* h      = agg + ND;                   // [N,D]
    float* dis    = h + ND;                     // [N]  (deg, then deg^-1/2)
    float* pooled = dis + GCN_N;                // [G,D]
    float* cnt    = pooled + (size_t)GCN_G * GCN_D;  // [G]

    const int nodeGrid = (GCN_N * 32 + 255) / 256;   // 6250
    const int edgeGrid = (GCN_E * 32 + 255) / 256;   // 80000

    // degree -> dis (in place)
    zero_kernel<<<(GCN_N + 255) / 256, 256, 0, stream>>>(dis, GCN_N);
    count_deg_kernel<<<(GCN_E + 255) / 256, 256, 0, stream>>>(ei, dis, GCN_E);
    rsqrt_deg_kernel<<<(GCN_N + 255) / 256, 256, 0, stream>>>(dis, GCN_N);

    // pooled / cnt init (must be re-zeroed every call)
    zero_kernel<<<((GCN_G * GCN_D + GCN_G) + 255) / 256, 256, 0, stream>>>(
        pooled, GCN_G * GCN_D + GCN_G);

    for (int i = 0; i < GCN_L; ++i) {
        const float* hin = (i == 0) ? x : h;
        const float* Wl  = W + (size_t)i * GCN_D * GCN_D;

        gemm_wmma_kernel<<<GCN_N / M_TILE, 256, 0, stream>>>(hin, Wl, t);
        init_self_kernel<<<nodeGrid, 256, 0, stream>>>(t, dis, b + (size_t)i * GCN_D,
                                                       agg, GCN_N);
        edge_scatter_kernel<<<edgeGrid, 256, 0, stream>>>(ei, t, dis, agg, GCN_E);
        bn_relu_kernel<<<nodeGrid, 256, 0, stream>>>(agg,
                                                     gamma + (size_t)i * GCN_D,
                                                     beta  + (size_t)i * GCN_D,
                                                     rmean + (size_t)i * GCN_D,
                                                     rvar  + (size_t)i * GCN_D,
                                                     h, GCN_N);
    }

    pool_add_kernel<<<nodeGrid, 256, 0, stream>>>(h, batch, pooled, cnt, GCN_N);
    out_gemm_kernel<<<(GCN_G * GCN_OUT + 255) / 256, 256, 0, stream>>>(
        pooled, cnt, W_out, b_out, out, GCN_G * GCN_OUT);
}